// Model_51659866636815
// MI455X (gfx1250) — compile-verified
//
#include <hip/hip_runtime.h>
#include <stdint.h>

// Problem constants (b=2, h=16, L=4096, d=16, dv=64, chunk=256)
#define NB   2
#define NH   16
#define SL   4096
#define DK   16
#define DV   64
#define CS   256
#define NC   (SL / CS)      // 16 chunks
#define BH   (NB * NH)      // 32

// Per-(bh,chunk) state record layout in workspace (floats)
#define OFF_SK   0           // Sum k            [16]
#define OFF_SKK  16          // Sum k (x) k      [256]
#define OFF_SV   272         // Sum v            [64]
#define OFF_KV   336         // K^T V            [16 x 64]
#define OFF_KKV  1360        // (k(x)k)^T V      [256 x 64]
#define REC      17744       // total floats per record (~35 MB workspace total)

typedef float v2f __attribute__((ext_vector_type(2)));
typedef float v8f __attribute__((ext_vector_type(8)));

// D = A(16x4) * B(4x16) + C, full f32 (exact w.r.t. reference precision)
__device__ __forceinline__ v8f wmma4(v2f a, v2f b, v8f c) {
  return __builtin_amdgcn_wmma_f32_16x16x4_f32(false, a, false, b, (short)0, c,
                                               false, false);
}

// LDS offset of a generic pointer to __shared__ (shared aperture is in the
// high 32 bits on gfx1250; low 32 bits are the LDS byte offset).
__device__ __forceinline__ uint32_t lds_off(const void* p) {
  return (uint32_t)(uintptr_t)p;
}

// Async DMA: copies 16 bytes from global address gbase + goff to LDS byte address lds_byte, ASYNCcnt-tracked.
__device__ __forceinline__ void async_b128(uint32_t lds_byte, const void* gbase,
                                           uint32_t goff) {
  asm volatile("global_load_async_to_lds_b128 %0, %1, %2"
               :: "v"(lds_byte), "v"(goff), "s"(gbase)
               : "memory");
}

__device__ __forceinline__ void wait_async0() {
  asm volatile("s_wait_asynccnt 0" ::: "memory");
}

// -------------------------------------------------------------------------
// Kernel 1: per-chunk local states. Dominant KKV matmul on the WMMA pipe.
// grid = BH*NC blocks, 256 threads (8 waves).
// -------------------------------------------------------------------------
__global__ __launch_bounds__(256) void k1_states(const float* __restrict__ K,
                                                 const float* __restrict__ V,
                                                 float* __restrict__ ws) {
  __shared__ __attribute__((aligned(16))) float lk[CS * DK];   // 16 KB
  __shared__ __attribute__((aligned(16))) float lv[CS * DV];   // 64 KB
  const int blk = blockIdx.x;
  const int bh  = blk / NC;
  const int cc  = blk % NC;
  const float* kg = K + ((size_t)bh * SL + (size_t)cc * CS) * DK;
  const float* vg = V + ((size_t)bh * SL + (size_t)cc * CS) * DV;
  const int tid = threadIdx.x;

  // Stage K,V tiles straight into LDS on the async path (no VGPR round-trip).
  {
    const uint32_t lkb = lds_off(lk), lvb = lds_off(lv);
    for (int i = tid; i < CS * DK / 4; i += 256)
      async_b128(lkb + i * 16, kg, i * 16);
    for (int i = tid; i < CS * DV / 4; i += 256)
      async_b128(lvb + i * 16, vg, i * 16);
  }
  wait_async0();
  __syncthreads();

  float* rec = ws + (size_t)blk * REC;

  // Small reductions (VALU; negligible vs matmuls)
  if (tid < 16) {
    float s = 0.f;
    for (int r = 0; r < CS; ++r) s += lk[r * DK + tid];
    rec[OFF_SK + tid] = s;
  }
  if (tid < 64) {
    float s = 0.f;
    for (int r = 0; r < CS; ++r) s += lv[r * DV + tid];
    rec[OFF_SV + tid] = s;
  }
  {
    const int d = tid >> 4, e = tid & 15;
    float s = 0.f;
    for (int r = 0; r < CS; ++r) s += lk[r * DK + d] * lk[r * DK + e];
    rec[OFF_SKK + tid] = s;
  }
  for (int i = 0; i < 4; ++i) {
    const int idx = tid + 256 * i;
    const int d = idx >> 6, f = idx & 63;
    float s = 0.f;
    for (int r = 0; r < CS; ++r) s += lk[r * DK + d] * lv[r * DV + f];
    rec[OFF_KV + idx] = s;
  }

  // KKV[de, f] = sum_c k[c,d]*k[c,e]*v[c,f]  -> 256x64 output, K-depth 256.
  // 64 tiles of 16x16; 8 waves x 8 tiles; uniform EXEC for WMMA.
  const int wave = tid >> 5, lane = tid & 31;
  const int half = lane >> 4, lm = lane & 15;
  for (int tt = wave; tt < 64; tt += 8) {
    const int tr = tt >> 2, tc = tt & 3;
    const int de = tr * 16 + lm;          // A-matrix row M for this lane
    const int d = de >> 4, e = de & 15;
    v8f acc = {};
    for (int kc = 0; kc < 64; ++kc) {
      const int c0 = 4 * kc + 2 * half;   // K index pair {c0, c0+1}
      v2f a, b;
      a.x = lk[c0 * DK + d] * lk[c0 * DK + e];
      a.y = lk[(c0 + 1) * DK + d] * lk[(c0 + 1) * DK + e];
      b.x = lv[c0 * DV + tc * 16 + lm];
      b.y = lv[(c0 + 1) * DV + tc * 16 + lm];
      acc = wmma4(a, b, acc);
    }
    for (int r = 0; r < 8; ++r) {
      const int m = r + 8 * half;         // D layout: vgpr r, lane half
      rec[OFF_KKV + (tr * 16 + m) * DV + tc * 16 + lm] = acc[r];
    }
  }
}

// -------------------------------------------------------------------------
// Kernel 2: exclusive prefix-scan of state records over chunks (in place).
// -------------------------------------------------------------------------
__global__ __launch_bounds__(256) void k2_scan(float* __restrict__ ws) {
  const int bh = blockIdx.x;
  for (int e = threadIdx.x; e < REC; e += 256) {
    float run = 0.f;
    for (int c = 0; c < NC; ++c) {
      const size_t idx = ((size_t)(bh * NC + c)) * REC + e;
      const float t = ws[idx];
      ws[idx] = run;
      run += t;
    }
  }
}

// -------------------------------------------------------------------------
// Kernel 3: outputs. 16 waves; wave w owns queries [16w,16w+16).
// acc[0..3] = output f-tiles, acc[4] = z (ones / broadcast-state B tiles).
// q is staged RAW; the d^-1/2 = 0.25 scale is folded into the A operands
// (for the quadratic term it folds into the 0.5 constant -> 0.03125).
// -------------------------------------------------------------------------
__global__ __launch_bounds__(512) void k3_output(const float* __restrict__ Q,
                                                 const float* __restrict__ K,
                                                 const float* __restrict__ V,
                                                 const float* __restrict__ ws,
                                                 float* __restrict__ out) {
  __shared__ __attribute__((aligned(16))) float lq[CS * DK];     // 16 KB (raw q)
  __shared__ __attribute__((aligned(16))) float lk[CS * DK];     // 16 KB
  __shared__ __attribute__((aligned(16))) float lv[CS * DV];     // 64 KB
  __shared__ __attribute__((aligned(16))) float lkkv[256 * DV];  // 64 KB
  __shared__ __attribute__((aligned(16))) float lkv[DK * DV];    // 4 KB
  __shared__ __attribute__((aligned(16))) float lskk[256];
  __shared__ __attribute__((aligned(16))) float lsk[16];
  __shared__ __attribute__((aligned(16))) float lsv[64];
  __shared__ __attribute__((aligned(16))) float lw[16 * 256];    // 16 KB, per-wave w tile

  const int blk = blockIdx.x;
  const int bh  = blk / NC;
  const int cc  = blk % NC;
  const int tid = threadIdx.x;
  const float* qg  = Q + ((size_t)bh * SL + (size_t)cc * CS) * DK;
  const float* kg  = K + ((size_t)bh * SL + (size_t)cc * CS) * DK;
  const float* vg  = V + ((size_t)bh * SL + (size_t)cc * CS) * DV;
  const float* rec = ws + (size_t)blk * REC;

  // Async DMA staging of all tiles and states (ASYNCcnt path).
  {
    const uint32_t lqb = lds_off(lq), lkb = lds_off(lk), lvb = lds_off(lv);
    const uint32_t lkkvb = lds_off(lkkv), lkvb = lds_off(lkv);
    const uint32_t lskkb = lds_off(lskk), lskb = lds_off(lsk), lsvb = lds_off(lsv);
    for (int i = tid; i < CS * DK / 4; i += 512) {
      async_b128(lqb + i * 16, qg, i * 16);
      async_b128(lkb + i * 16, kg, i * 16);
    }
    for (int i = tid; i < CS * DV / 4; i += 512)
      async_b128(lvb + i * 16, vg, i * 16);
    for (int i = tid; i < 256 * DV / 4; i += 512)
      async_b128(lkkvb + i * 16, rec + OFF_KKV, i * 16);
    if (tid < DK * DV / 4) async_b128(lkvb + tid * 16, rec + OFF_KV, tid * 16);
    if (tid < 64)          async_b128(lskkb + tid * 16, rec + OFF_SKK, tid * 16);
    if (tid < 16)          async_b128(lsvb + tid * 16, rec + OFF_SV, tid * 16);
    if (tid < 4)           async_b128(lskb + tid * 16, rec + OFF_SK, tid * 16);
  }
  wait_async0();
  __syncthreads();

  const int wave = tid >> 5, lane = tid & 31;
  const int half = lane >> 4, lm = lane & 15;
  const int qrow = wave * 16 + lm;       // A-matrix row (query) for this lane
  const float QS = 0.25f;                // d^-0.5

  v8f acc[5] = {};

  // ---- inter-chunk linear: (QS*q) @ KV  (+ z via broadcast Sk tile) ----
  #pragma unroll
  for (int kc = 0; kc < 4; ++kc) {
    const int k0 = 4 * kc + 2 * half;
    v2f a = *(const v2f*)&lq[qrow * DK + k0];
    a.x *= QS; a.y *= QS;
    #pragma unroll
    for (int t = 0; t < 4; ++t) {
      v2f b;
      b.x = lkv[k0 * DV + t * 16 + lm];
      b.y = lkv[(k0 + 1) * DV + t * 16 + lm];
      acc[t] = wmma4(a, b, acc[t]);
    }
    v2f bz; bz.x = lsk[k0]; bz.y = lsk[k0 + 1];
    acc[4] = wmma4(a, bz, acc[4]);
  }

  // ---- inter-chunk quadratic: 0.5*QS^2*(q (x) q) @ KKV (+ z via SKK) ----
  for (int kc = 0; kc < 64; ++kc) {
    const int de0 = 4 * kc + 2 * half;
    const int de1 = de0 + 1;
    v2f a;
    a.x = 0.03125f * lq[qrow * DK + (de0 >> 4)] * lq[qrow * DK + (de0 & 15)];
    a.y = 0.03125f * lq[qrow * DK + (de1 >> 4)] * lq[qrow * DK + (de1 & 15)];
    #pragma unroll
    for (int t = 0; t < 4; ++t) {
      v2f b;
      b.x = lkkv[de0 * DV + t * 16 + lm];
      b.y = lkkv[de1 * DV + t * 16 + lm];
      acc[t] = wmma4(a, b, acc[t]);
    }
    v2f bz; bz.x = lskk[de0]; bz.y = lskk[de1];
    acc[4] = wmma4(a, bz, acc[4]);
  }

  // ---- intra-chunk: S = (QS*Q) K^T ; w = 1 + s + s^2/2 (causal) ; o += w V --
  float* wbuf = lw + wave * 256;
  for (int j = 0; j <= wave; ++j) {      // wave-uniform bound -> EXEC full
    v8f s = {};
    #pragma unroll
    for (int kc = 0; kc < 4; ++kc) {
      const int k0 = 4 * kc + 2 * half;
      v2f a = *(const v2f*)&lq[qrow * DK + k0];
      a.x *= QS; a.y *= QS;
      v2f b = *(const v2f*)&lk[(j * 16 + lm) * DK + k0];
      s = wmma4(a, b, s);
    }
    // Stage w tile (D layout -> LDS row-major) with nonlinearity + mask
    #pragma unroll
    for (int r = 0; r < 8; ++r) {
      const int m = r + 8 * half;                 // query row in strip
      const float sv = s[r];
      float wv = 1.f + sv + 0.5f * sv * sv;       // includes zeroth-order "1"
      wv = (j == wave && lm > m) ? 0.f : wv;      // causal on diagonal tile
      wbuf[m * 16 + lm] = wv;
    }
    asm volatile("s_wait_dscnt 0" ::: "memory");  // wave-internal LDS RAW
    // o += w @ V(rows of tile j); z += w @ ones
    #pragma unroll
    for (int kc = 0; kc < 4; ++kc) {
      const int k0 = 4 * kc + 2 * half;
      v2f a = *(const v2f*)&wbuf[lm * 16 + k0];
      #pragma unroll
      for (int t = 0; t < 4; ++t) {
        v2f b;
        b.x = lv[(j * 16 + k0) * DV + t * 16 + lm];
        b.y = lv[(j * 16 + k0 + 1) * DV + t * 16 + lm];
        acc[t] = wmma4(a, b, acc[t]);
      }
      v2f bone; bone.x = 1.f; bone.y = 1.f;
      acc[4] = wmma4(a, bone, acc[4]);
    }
  }

  // ---- finalize: out = (o + Sv) / (z + count + 1e-6) ----
  const float zbase = (float)(cc * CS) + 1e-6f;   // ones from previous chunks
  #pragma unroll
  for (int r = 0; r < 8; ++r) {
    const int m = r + 8 * half;
    const float rz = 1.f / (acc[4][r] + zbase);
    const size_t row = (size_t)bh * SL + (size_t)cc * CS + (size_t)(wave * 16 + m);
    #pragma unroll
    for (int t = 0; t < 4; ++t) {
      out[row * DV + t * 16 + lm] = (acc[t][r] + lsv[t * 16 + lm]) * rz;
    }
  }
}

// -------------------------------------------------------------------------
extern "C" void kernel_launch(void* const* d_in, const int* in_sizes, int n_in,
                              void* d_out, int out_size, void* d_ws, size_t ws_size,
                              hipStream_t stream) {
  (void)in_sizes; (void)n_in; (void)out_size; (void)ws_size;
  const float* q = (const float*)d_in[0];
  const float* k = (const float*)d_in[1];
  const float* v = (const float*)d_in[2];
  float* out = (float*)d_out;
  float* ws  = (float*)d_ws;   // needs BH*NC*REC*4 ~= 35 MB

  k1_states<<<dim3(BH * NC), dim3(256), 0, stream>>>(k, v, ws);
  k2_scan  <<<dim3(BH),      dim3(256), 0, stream>>>(ws);
  k3_output<<<dim3(BH * NC), dim3(512), 0, stream>>>(q, k, v, ws, out);
}